// Memory_11905649344740
// MI455X (gfx1250) — compile-verified
//
#include <hip/hip_runtime.h>

namespace {

constexpr int B_ = 128, M_ = 4096, U_ = 64, D_ = 32;
constexpr float NEG_INF = -1e30f;

typedef __attribute__((ext_vector_type(2))) float v2f;
typedef __attribute__((ext_vector_type(8))) float v8f;

// D(16x16,f32) = A(16x4,f32) * B(4x16,f32) + C ; chained over K.
__device__ inline v8f wmma_k4(v2f a, v2f b, v8f c) {
  return __builtin_amdgcn_wmma_f32_16x16x4_f32(
      /*neg_a=*/false, a, /*neg_b=*/false, b,
      /*c_mod=*/(short)0, c, /*reuse_a=*/false, /*reuse_b=*/false);
}

__device__ inline v8f zero8() {
  v8f c;
#pragma unroll
  for (int i = 0; i < 8; ++i) c[i] = 0.0f;
  return c;
}

// online softmax combine: (mx,sm) <- merge with (om,os)
__device__ inline void combine(float& mx, float& sm, float om, float os) {
  float nm = fmaxf(mx, om);
  sm = sm * __expf(mx - nm) + os * __expf(om - nm);
  mx = nm;
}

// Pass 1: per-(b,u) running max and sum(exp) over all m. 1 wave per block.
__global__ __launch_bounds__(32) void k_stats(
    const float* __restrict__ att, const float* __restrict__ matt,
    const float* __restrict__ temp, const unsigned char* __restrict__ mask,
    float* __restrict__ maxbuf, float* __restrict__ sumbuf) {
  const int b0 = blockIdx.x * 16;
  const int u = blockIdx.y;
  const int lane = threadIdx.x;
  const int row = lane & 15, hi = lane >> 4;
  const float inv_t = 1.0f / temp[u];

  // A fragments: A[b_local][d], kept in registers for the whole m loop.
  v2f a[8];
  {
    const float* pa = att + (size_t)(b0 + row) * U_ * D_ + u * D_ + 2 * hi;
#pragma unroll
    for (int kk = 0; kk < 8; ++kk) a[kk] = *(const v2f*)(pa + 4 * kk);
  }

  float mx[8], sm[8];
#pragma unroll
  for (int r = 0; r < 8; ++r) { mx[r] = -3.0e38f; sm[r] = 0.0f; }

  for (int m0 = 0; m0 < M_; m0 += 16) {
    const float* pb = matt + (size_t)(m0 + row) * U_ * D_ + u * D_ + 2 * hi;
    v8f c = zero8();
#pragma unroll
    for (int kk = 0; kk < 8; ++kk)
      c = wmma_k4(a[kk], *(const v2f*)(pb + 4 * kk), c);

    const int m = m0 + row;  // this lane's m column of the C tile
#pragma unroll
    for (int r = 0; r < 8; ++r) {
      const int b = b0 + r + 8 * hi;
      const unsigned char mk = mask[(size_t)b * M_ * U_ + (size_t)m * U_ + u];
      float s = (mk ? NEG_INF : c[r]) * inv_t;
      combine(mx[r], sm[r], s, 1.0f);
    }
  }
  // reduce across the 16 lanes of each half (same 8 b's, disjoint m's)
#pragma unroll
  for (int off = 1; off < 16; off <<= 1) {
#pragma unroll
    for (int r = 0; r < 8; ++r) {
      float om = __shfl_xor(mx[r], off, 32);
      float os = __shfl_xor(sm[r], off, 32);
      combine(mx[r], sm[r], om, os);
    }
  }
  if (row == 0) {
#pragma unroll
    for (int r = 0; r < 8; ++r) {
      const int b = b0 + r + 8 * hi;
      maxbuf[b * U_ + u] = mx[r];
      sumbuf[b * U_ + u] = sm[r];
    }
  }
}

// Pass 2: recompute scores, normalize -> weights (u-contiguous stores via LDS),
// and accumulate outputs with a second WMMA GEMM. 32 waves = 32 u's per block.
__global__ __launch_bounds__(1024) void k_main(
    const float* __restrict__ att, const float* __restrict__ matt,
    const float* __restrict__ mem, const float* __restrict__ temp,
    const unsigned char* __restrict__ mask,
    const float* __restrict__ maxbuf, const float* __restrict__ sumbuf,
    float* __restrict__ weights, float* __restrict__ outputs) {
  // [b_local*16 + m_local][u] with pitch 36 (bank spread + float4 alignment)
  __shared__ float lds_w[16 * 16 * 36];

  const int b0 = blockIdx.x * 16;
  const int u0 = blockIdx.y * 32;
  const int tid = threadIdx.x;
  const int w = tid >> 5;  // local u == wave id
  const int lane = tid & 31;
  const int row = lane & 15, hi = lane >> 4;
  const int u = u0 + w;
  const float inv_t = 1.0f / temp[u];

  v2f a[8];
  {
    const float* pa = att + (size_t)(b0 + row) * U_ * D_ + u * D_ + 2 * hi;
#pragma unroll
    for (int kk = 0; kk < 8; ++kk) a[kk] = *(const v2f*)(pa + 4 * kk);
  }
  float mx[8], rs[8];
#pragma unroll
  for (int r = 0; r < 8; ++r) {
    const int b = b0 + r + 8 * hi;
    mx[r] = maxbuf[b * U_ + u];
    rs[r] = 1.0f / sumbuf[b * U_ + u];
  }
  v8f acc0 = zero8(), acc1 = zero8();  // out tile, d = 0..15 and 16..31

  const int p = tid >> 2;  // (b_local*16 + m_local) for cooperative store
  const int q = tid & 3;   // 8-u chunk index

  for (int m0 = 0; m0 < M_; m0 += 16) {
    // ---- einsum 1: scores(16b x 16m), K = D = 32 ----
    const float* pb = matt + (size_t)(m0 + row) * U_ * D_ + u * D_ + 2 * hi;
    v8f c = zero8();
#pragma unroll
    for (int kk = 0; kk < 8; ++kk)
      c = wmma_k4(a[kk], *(const v2f*)(pb + 4 * kk), c);

    const int m = m0 + row;
#pragma unroll
    for (int r = 0; r < 8; ++r) {
      const int b = b0 + r + 8 * hi;
      const unsigned char mk = mask[(size_t)b * M_ * U_ + (size_t)m * U_ + u];
      float s = (mk ? NEG_INF : c[r]) * inv_t;
      float wgt = __expf(s - mx[r]) * rs[r];
      lds_w[((r + 8 * hi) * 16 + row) * 36 + w] = wgt;
    }
    __syncthreads();

    // ---- coalesced u-contiguous store of the 16x16x32 weight tile ----
    {
      float* dst = weights + (size_t)(b0 + (p >> 4)) * M_ * U_ +
                   (size_t)(m0 + (p & 15)) * U_ + u0 + q * 8;
      const float* src = lds_w + p * 36 + q * 8;
      *(float4*)(dst) = *(const float4*)(src);
      *(float4*)(dst + 4) = *(const float4*)(src + 4);
    }

    // ---- einsum 2: out(16b x 32d) += W(16b x 16m) * memory(16m x 32d) ----
#pragma unroll
    for (int kk = 0; kk < 4; ++kk) {
      const int k = 4 * kk + 2 * hi;
      v2f a2;
      a2[0] = lds_w[(row * 16 + k) * 36 + w];
      a2[1] = lds_w[(row * 16 + k + 1) * 36 + w];
      const float* pm0 = mem + (size_t)(m0 + k) * U_ * D_ + u * D_;
      const float* pm1 = pm0 + U_ * D_;
      v2f b2l, b2h;
      b2l[0] = pm0[row];      b2l[1] = pm1[row];
      b2h[0] = pm0[16 + row]; b2h[1] = pm1[16 + row];
      acc0 = wmma_k4(a2, b2l, acc0);
      acc1 = wmma_k4(a2, b2h, acc1);
    }
    __syncthreads();
  }

  // outputs[b, u, d]
#pragma unroll
  for (int r = 0; r < 8; ++r) {
    const int b = b0 + r + 8 * hi;
    float* po = outputs + (size_t)b * U_ * D_ + u * D_;
    po[row] = acc0[r];
    po[16 + row] = acc1[r];
  }
}

__global__ void k_copy(const float4* __restrict__ src, float4* __restrict__ dst,
                       int n4) {
  int i = blockIdx.x * blockDim.x + threadIdx.x;
  const int stride = gridDim.x * blockDim.x;
  for (; i < n4; i += stride) dst[i] = src[i];
}

}  // namespace

extern "C" void kernel_launch(void* const* d_in, const int* in_sizes, int n_in,
                              void* d_out, int out_size, void* d_ws,
                              size_t ws_size, hipStream_t stream) {
  (void)in_sizes; (void)n_in; (void)out_size; (void)ws_size;
  const float* att = (const float*)d_in[0];          // [B,U,D]
  const float* matt = (const float*)d_in[1];         // [M,U,D]
  const float* mem = (const float*)d_in[2];          // [M,U,D]
  const float* temp = (const float*)d_in[3];         // [U]
  const unsigned char* mask = (const unsigned char*)d_in[4];  // [B,M,U] bool

  float* outputs = (float*)d_out;                    // B*U*D
  float* weights = outputs + (size_t)B_ * U_ * D_;   // B*M*U
  float* memcopy = weights + (size_t)B_ * M_ * U_;   // M*U*D

  float* maxbuf = (float*)d_ws;          // B*U floats
  float* sumbuf = maxbuf + B_ * U_;      // B*U floats

  k_stats<<<dim3(B_ / 16, U_), 32, 0, stream>>>(att, matt, temp, mask, maxbuf,
                                                sumbuf);
  k_main<<<dim3(B_ / 16, U_ / 32), 1024, 0, stream>>>(
      att, matt, mem, temp, mask, maxbuf, sumbuf, weights, outputs);

  const int n4 = (M_ * U_ * D_) / 4;
  k_copy<<<512, 256, 0, stream>>>((const float4*)mem, (float4*)memcopy, n4);
}